// OneByOneMerge3_2D_20538533609781
// MI455X (gfx1250) — compile-verified
//
#include <hip/hip_runtime.h>
#include <stdint.h>

// out[..., c] = t1[...,c]*w[c,0] + t2[...,c]*w[c,1] + t3[...,c]*w[c,2]
// Shapes: t* = (16,128,128,256) f32, weights = (256,3) f32.
// Pure HBM-bound streaming kernel: 1 GiB traffic, ~44us floor @ 23.3 TB/s.
// No WMMA on purpose (0.31 FLOP/byte); CDNA5 paths used: async global->LDS
// weight staging (ASYNCcnt) + non-temporal b128 stream loads/stores.

typedef float v4f __attribute__((ext_vector_type(4)));

__global__ __launch_bounds__(256) void merge3_nhwc_kernel(
    const float* __restrict__ t1, const float* __restrict__ t2,
    const float* __restrict__ t3, const float* __restrict__ w,
    float* __restrict__ out, long long n4)
{
    // 256 channels * 3 weights = 768 f32 = 3 KB in LDS.
    __shared__ float sw[768];

    const int tid = threadIdx.x;

    // --- Stage weight table into LDS via CDNA5 async copy ------------------
    // Threads 0..191 each move one 16B chunk: GLOBAL_LOAD_ASYNC_TO_LDS_B128.
    if (tid < 192) {
        uint32_t lds_off = (uint32_t)(uintptr_t)(&sw[0]) + (uint32_t)tid * 16u;
        uint64_t gaddr   = (uint64_t)(uintptr_t)w + (uint64_t)tid * 16u;
        asm volatile("global_load_async_to_lds_b128 %0, %1, off"
                     :: "v"(lds_off), "v"(gaddr)
                     : "memory");
    }
    asm volatile("s_wait_asynccnt 0" ::: "memory");
    __syncthreads();

    // --- Per-thread loop-invariant weights ---------------------------------
    // blockDim=256, 4 floats/thread => thread's channel group = (tid & 63),
    // channels c0 = 4*(tid&63)..c0+3. Weight layout is w[c*3 + j], so the 12
    // floats this thread needs are contiguous & 16B-aligned: 3x ds_load_b128.
    const int cg = tid & 63;
    const float* wp = &sw[cg * 12];
    const v4f wa = *(const v4f*)(wp + 0);
    const v4f wb = *(const v4f*)(wp + 4);
    const v4f wc = *(const v4f*)(wp + 8);
    // Unpack (w0,w1,w2) for each of the 4 channels.
    const float w0x = wa.x, w1x = wa.y, w2x = wa.z;   // channel c0+0
    const float w0y = wa.w, w1y = wb.x, w2y = wb.y;   // channel c0+1
    const float w0z = wb.z, w1z = wb.w, w2z = wc.x;   // channel c0+2
    const float w0w = wc.y, w1w = wc.z, w2w = wc.w;   // channel c0+3

    // --- Streaming main loop: non-temporal b128 traffic --------------------
    const v4f* a = (const v4f*)t1;
    const v4f* b = (const v4f*)t2;
    const v4f* c = (const v4f*)t3;
    v4f*       o = (v4f*)out;

    const long long stride = (long long)gridDim.x * 256;  // multiple of 64
    for (long long i = (long long)blockIdx.x * 256 + tid; i < n4; i += stride) {
        v4f x = __builtin_nontemporal_load(a + i);
        v4f y = __builtin_nontemporal_load(b + i);
        v4f z = __builtin_nontemporal_load(c + i);
        v4f r;
        r.x = __builtin_fmaf(x.x, w0x, __builtin_fmaf(y.x, w1x, z.x * w2x));
        r.y = __builtin_fmaf(x.y, w0y, __builtin_fmaf(y.y, w1y, z.y * w2y));
        r.z = __builtin_fmaf(x.z, w0z, __builtin_fmaf(y.z, w1z, z.z * w2z));
        r.w = __builtin_fmaf(x.w, w0w, __builtin_fmaf(y.w, w1w, z.w * w2w));
        __builtin_nontemporal_store(r, o + i);
    }
}

extern "C" void kernel_launch(void* const* d_in, const int* in_sizes, int n_in,
                              void* d_out, int out_size, void* d_ws, size_t ws_size,
                              hipStream_t stream) {
    const float* t1 = (const float*)d_in[0];
    const float* t2 = (const float*)d_in[1];
    const float* t3 = (const float*)d_in[2];
    const float* w  = (const float*)d_in[3];
    float* out = (float*)d_out;

    const long long n4 = (long long)out_size / 4;   // float4 groups (16 Mi)

    // Grid-stride: ~2M threads (8192 blocks x 8 waves), ~8 float4 iters each.
    long long want = (n4 + 255) / 256;
    int blocks = (int)(want < 8192 ? want : 8192);
    if (blocks < 1) blocks = 1;

    merge3_nhwc_kernel<<<blocks, 256, 0, stream>>>(t1, t2, t3, w, out, n4);
}